// NLLLogisticHazardLoss_75634374083217
// MI455X (gfx1250) — compile-verified
//
#include <hip/hip_runtime.h>
#include <hip/hip_bf16.h>

typedef __attribute__((ext_vector_type(2))) float v2f;
typedef __attribute__((ext_vector_type(4))) float v4f;
typedef __attribute__((ext_vector_type(8))) float v8f;

#define TBINS 64
#define LOSS_EPS 1e-9f
#define F_LOG2E 1.44269504088896340736f
#define F_LN2   0.69314718055994530942f

__global__ void hz_init_ws(float* ws) {
    ws[0] = 0.0f;
    ws[1] = 0.0f;
}

__global__ __launch_bounds__(256) void hz_loss_kernel(
    const float* __restrict__ preds,      // [N, 64]
    const float* __restrict__ weight,     // [64]
    const float* __restrict__ sw,         // [N]
    const int*   __restrict__ durations,  // [N]
    const int*   __restrict__ events,     // [N]
    float* __restrict__ ws,               // [num, den] accumulators
    int nRows)
{
    __shared__ float wlds[TBINS];
    const int tid = threadIdx.x;
    if (tid < TBINS) wlds[tid] = weight[tid];
    __syncthreads();

    const int lane  = tid & 31;          // wave32
    const int hal   = lane >> 4;         // half-wave: K={0,1} vs K={2,3}
    const int laneM = lane & 15;         // row within 16-row tile
    const int wavesPerBlock = blockDim.x >> 5;
    const int waveId     = blockIdx.x * wavesPerBlock + (tid >> 5);
    const int totalWaves = gridDim.x * wavesPerBlock;
    const int nTiles = nRows >> 4;       // 16 rows per WMMA tile

    // Hoist bin weights into the B operand once per wave: B[k][n] = w[4k+K].
    // B layout mirrors A: VGPR0 holds rows K=0 (lanes 0-15) / K=2 (lanes 16-31),
    // VGPR1 holds K=1 / K=3 -> same t indices this lane uses for A.
    v2f bW[16];
    #pragma unroll
    for (int k = 0; k < 16; ++k) {
        const int t0 = 4 * k + 2 * hal;
        bW[k].x = wlds[t0];
        bW[k].y = wlds[t0 + 1];
    }

    const float laneMask = (laneM == 0) ? 1.0f : 0.0f;  // commit lanes 0 & 16 only
    float fnum = 0.0f, fden = 0.0f;

    for (int tile = waveId; tile < nTiles; tile += totalWaves) {
        const int base = tile << 4;
        const int row  = base + laneM;

        // prefetch next tile's pred rows (global_prefetch_b8)
        if (tile + totalWaves < nTiles) {
            const int nrow = ((tile + totalWaves) << 4) + laneM;
            __builtin_prefetch(preds + (size_t)nrow * TBINS, 0, 3);
        }

        int d = durations[row];
        d = (d < 0) ? 0 : ((d > TBINS - 1) ? TBINS - 1 : d);
        const int de = (events[row] != 0) ? d : -1;   // t==de -> event bin
        const float* prow = preds + (size_t)row * TBINS;

        // Software pipeline: issue all 16 independent NT b64 loads up front
        // (128 B/lane in flight, one wait), then compute back-to-back.
        v2f xs[16];
        #pragma unroll
        for (int k = 0; k < 16; ++k) {
            xs[k] = __builtin_nontemporal_load((const v2f*)(prow + 4 * k + 2 * hal));
        }

        v8f acc = {};  // 16x16 f32 accumulator: weighted row sums (cols identical)
        #pragma unroll
        for (int k = 0; k < 16; ++k) {
            const int t0 = 4 * k + 2 * hal;
            v2f a;
            #pragma unroll
            for (int j = 0; j < 2; ++j) {
                const int   t  = t0 + j;
                const float xv = (j == 0) ? xs[k].x : xs[k].y;
                // softplus(x) = max(x,0) + ln2*log2(1 + 2^(-|x|*log2e))
                // bare v_exp_f32 / v_log_f32: arg of log is in (1,2], no fixups
                const float e2 = __builtin_amdgcn_exp2f(-fabsf(xv) * F_LOG2E);
                const float lg = __builtin_amdgcn_logf(1.0f + e2);
                const float sp = __builtin_fmaf(lg, F_LN2, fmaxf(xv, 0.0f));
                // t<d: softplus(x); t==d: event ? softplus(x)-x (= -log h) : softplus(x); t>d: 0
                float av = (t == de) ? (sp - xv) : sp;
                av = (t <= d) ? av : 0.0f;
                if (j == 0) a.x = av; else a.y = av;
            }
            // D(16x16,f32) += A(16x4) x B(4x16,=weights): weighted per-row sums
            acc = __builtin_amdgcn_wmma_f32_16x16x4_f32(
                false, a, false, bW[k], (short)0, acc, false, false);
        }

        // acc[v] on every lane of half `hal` = weighted rowsum(base + v + 8*hal)
        const float* swp = sw + base + hal * 8;
        const v4f s0 = *(const v4f*)(swp);
        const v4f s1 = *(const v4f*)(swp + 4);
        const float partial =
            acc[0] * s0.x + acc[1] * s0.y + acc[2] * s0.z + acc[3] * s0.w +
            acc[4] * s1.x + acc[5] * s1.y + acc[6] * s1.z + acc[7] * s1.w;
        const float denp = s0.x + s0.y + s0.z + s0.w + s1.x + s1.y + s1.z + s1.w;
        fnum += partial * laneMask;   // identical across each half-wave: mask to lanes 0,16
        fden += denp * laneMask;
    }

    // wave32 tree reduction
    #pragma unroll
    for (int off = 16; off > 0; off >>= 1) {
        fnum += __shfl_down(fnum, off, 32);
        fden += __shfl_down(fden, off, 32);
    }
    if (lane == 0) {
        atomicAdd(&ws[0], fnum);
        atomicAdd(&ws[1], fden);
    }
}

__global__ void hz_finalize(const float* __restrict__ ws, float* __restrict__ out) {
    out[0] = ws[0] / fmaxf(ws[1], LOSS_EPS);
}

extern "C" void kernel_launch(void* const* d_in, const int* in_sizes, int n_in,
                              void* d_out, int out_size, void* d_ws, size_t ws_size,
                              hipStream_t stream) {
    const float* preds     = (const float*)d_in[0];
    const float* weight    = (const float*)d_in[1];
    const float* sw        = (const float*)d_in[2];
    const int*   durations = (const int*)d_in[3];
    const int*   events    = (const int*)d_in[4];
    float* out = (float*)d_out;
    float* ws  = (float*)d_ws;

    const int nRows  = in_sizes[3];            // N from durations
    const int nTiles = nRows >> 4;             // 16 rows / tile
    const int wavesPerBlock = 256 / 32;        // 8 waves per block
    // 2 tiles per wave: amortizes setup and keeps the next-tile prefetch live
    int blocks = (nTiles / 2 + wavesPerBlock - 1) / wavesPerBlock;
    if (blocks < 1) blocks = 1;
    if (blocks > 16384) blocks = 16384;

    hz_init_ws<<<1, 1, 0, stream>>>(ws);
    hz_loss_kernel<<<blocks, 256, 0, stream>>>(preds, weight, sw, durations,
                                               events, ws, nRows);
    hz_finalize<<<1, 1, 0, stream>>>(ws, out);
}